// RotatedIoU3DLoss_64604898066824
// MI455X (gfx1250) — compile-verified
//
#include <hip/hip_runtime.h>
#include <math.h>

#define EPS      1e-8f
#define NCAND    24
#define THREADS  256
#define NBLOCKS  1024

typedef float v2f __attribute__((ext_vector_type(2)));
typedef float v8f __attribute__((ext_vector_type(8)));

// ---------------------------------------------------------------------------
// Wave32 sum using the CDNA5 matrix core.
// A (16x4 f32, 2 VGPRs): a[0] = x  -> A[m][0] = x[lane m], A[m][2] = x[lane m+16]
//                        a[1] = 0  -> K=1,3 rows are zero
// B (4x16 f32) = all ones.
// D[m][n] = x[m] + x[m+16] for every n.  C/D layout: lane n<16 vgpr v = D[v][n],
// lane n>=16 vgpr v = D[v+8][n-16].  Sum of the 8 D components gives half-sums;
// one xor-16 shuffle completes the full 32-lane sum (broadcast to all lanes).
// Requires EXEC == all ones (call only from reconverged full waves).
// ---------------------------------------------------------------------------
__device__ __forceinline__ float wave_sum32(float x) {
  v2f a; a[0] = x;    a[1] = 0.0f;
  v2f b; b[0] = 1.0f; b[1] = 1.0f;
  v8f c = {};
  v8f d = __builtin_amdgcn_wmma_f32_16x16x4_f32(false, a, false, b,
                                                (short)0, c, false, false);
  float p = ((d[0] + d[1]) + (d[2] + d[3])) + ((d[4] + d[5]) + (d[6] + d[7]));
  p += __shfl_xor(p, 16, 32);
  return p;
}

// ---------------------------------------------------------------------------
// Per box-pair rotated 3D IoU loss (fp32, mirrors the JAX reference exactly).
// ---------------------------------------------------------------------------
__device__ __forceinline__ float pair_loss(const float* __restrict__ bp,
                                           const float* __restrict__ bt,
                                           float wgt) {
  float b1[7], b2[7];
#pragma unroll
  for (int i = 0; i < 7; ++i) { b1[i] = bp[i]; b2[i] = bt[i]; }

  const float tx[4] = { 0.5f, -0.5f, -0.5f,  0.5f};
  const float ty[4] = { 0.5f,  0.5f, -0.5f, -0.5f};

  float c1x[4], c1y[4], c2x[4], c2y[4];
  {
    float ca = cosf(b1[6]), sa = sinf(b1[6]);
#pragma unroll
    for (int i = 0; i < 4; ++i) {
      float x4 = tx[i] * b1[3], y4 = ty[i] * b1[4];
      c1x[i] = x4 * ca - y4 * sa + b1[0];
      c1y[i] = x4 * sa + y4 * ca + b1[1];
    }
    float cb = cosf(b2[6]), sb = sinf(b2[6]);
#pragma unroll
    for (int i = 0; i < 4; ++i) {
      float x4 = tx[i] * b2[3], y4 = ty[i] * b2[4];
      c2x[i] = x4 * cb - y4 * sb + b2[0];
      c2y[i] = x4 * sb + y4 * cb + b2[1];
    }
  }

  float vx[NCAND], vy[NCAND];
  bool  vm[NCAND];

  // candidates 0..3: c1 corners, mask = c1-in-c2
  {
    float ax = c2x[0], ay = c2y[0];
    float abx = c2x[1] - ax, aby = c2y[1] - ay;
    float adx = c2x[3] - ax, ady = c2y[3] - ay;
    float dab = abx * abx + aby * aby;
    float dad = adx * adx + ady * ady;
    const float e = 1e-6f;
#pragma unroll
    for (int i = 0; i < 4; ++i) {
      float amx = c1x[i] - ax, amy = c1y[i] - ay;
      float pab = (amx * abx + amy * aby) / dab;
      float pad = (amx * adx + amy * ady) / dad;
      vm[i] = (pab > -e) && (pab < 1.0f + e) && (pad > -e) && (pad < 1.0f + e);
      vx[i] = c1x[i]; vy[i] = c1y[i];
    }
  }
  // candidates 4..7: c2 corners, mask = c2-in-c1
  {
    float ax = c1x[0], ay = c1y[0];
    float abx = c1x[1] - ax, aby = c1y[1] - ay;
    float adx = c1x[3] - ax, ady = c1y[3] - ay;
    float dab = abx * abx + aby * aby;
    float dad = adx * adx + ady * ady;
    const float e = 1e-6f;
#pragma unroll
    for (int i = 0; i < 4; ++i) {
      float amx = c2x[i] - ax, amy = c2y[i] - ay;
      float pab = (amx * abx + amy * aby) / dab;
      float pad = (amx * adx + amy * ady) / dad;
      vm[4 + i] = (pab > -e) && (pab < 1.0f + e) && (pad > -e) && (pad < 1.0f + e);
      vx[4 + i] = c2x[i]; vy[4 + i] = c2y[i];
    }
  }
  // candidates 8..23: the 16 edge-edge intersections
#pragma unroll
  for (int i = 0; i < 4; ++i) {
    int i1 = (i + 1) & 3;
    float p1x = c1x[i], p1y = c1y[i];
    float d1x = c1x[i1] - p1x, d1y = c1y[i1] - p1y;
#pragma unroll
    for (int j = 0; j < 4; ++j) {
      int j1 = (j + 1) & 3;
      float p3x = c2x[j], p3y = c2y[j];
      float d2x = c2x[j1] - p3x, d2y = c2y[j1] - p3y;
      float qpx = p3x - p1x, qpy = p3y - p1y;
      float den = (d1x * d2y - d1y * d2x) + EPS;
      float t = (qpx * d2y - qpy * d2x) / den;
      float u = (qpx * d1y - qpy * d1x) / den;
      bool  m = (t > 0.0f) && (t < 1.0f) && (u > 0.0f) && (u < 1.0f);
      int   k = 8 + i * 4 + j;
      vm[k] = m;
      vx[k] = m ? (p1x + t * d1x) : 0.0f;
      vy[k] = m ? (p1y + t * d1y) : 0.0f;
    }
  }

  // mean of valid vertices (num clipped to >= 1)
  int   cnt = 0;
  float sx = 0.0f, sy = 0.0f;
#pragma unroll
  for (int k = 0; k < NCAND; ++k) {
    cnt += vm[k] ? 1 : 0;
    sx  += vm[k] ? vx[k] : 0.0f;
    sy  += vm[k] ? vy[k] : 0.0f;
  }
  float num = (float)(cnt < 1 ? 1 : cnt);
  float mx = sx / num, my = sy / num;

  // angles (masked -> 1e9 so they sort last, as in the reference)
  float ang[NCAND];
#pragma unroll
  for (int k = 0; k < NCAND; ++k)
    ang[k] = vm[k] ? atan2f(vy[k] - my, vx[k] - mx) : 1.0e9f;

  // stable odd-even transposition sort by angle (fully unrolled, static idx)
#pragma unroll
  for (int r = 0; r < NCAND; ++r) {
#pragma unroll
    for (int i = (r & 1); i + 1 < NCAND; i += 2) {
      float a0 = ang[i], a1 = ang[i + 1];
      if (a0 > a1) {
        ang[i] = a1; ang[i + 1] = a0;
        float t0 = vx[i]; vx[i] = vx[i + 1]; vx[i + 1] = t0;
        float t1 = vy[i]; vy[i] = vy[i + 1]; vy[i + 1] = t1;
      }
    }
  }

  // vertices past the valid prefix collapse to the first sorted vertex
  float s0x = vx[0], s0y = vy[0];
  float px[NCAND], py[NCAND];
#pragma unroll
  for (int k = 0; k < NCAND; ++k) {
    bool v = k < cnt;
    px[k] = v ? vx[k] : s0x;
    py[k] = v ? vy[k] : s0y;
  }
  // shoelace
  float ssum = 0.0f;
#pragma unroll
  for (int k = 0; k < NCAND; ++k) {
    int k2 = (k + 1) % NCAND;
    ssum += px[k] * py[k2] - py[k] * px[k2];
  }
  float inter2d = 0.5f * fabsf(ssum);

  // 3D extension
  float zmax = fminf(b1[2] + 0.5f * b1[5], b2[2] + 0.5f * b2[5]);
  float zmin = fmaxf(b1[2] - 0.5f * b1[5], b2[2] - 0.5f * b2[5]);
  float inter3d = inter2d * fmaxf(zmax - zmin, 0.0f);
  float vol1 = b1[3] * b1[4] * b1[5];
  float vol2 = b2[3] * b2[4] * b2[5];
  float iou  = inter3d / (vol1 + vol2 - inter3d + EPS);
  return (1.0f - iou) * wgt;
}

// ---------------------------------------------------------------------------
// Pass 1: grid-stride compute + deterministic block reduction -> partials
// ---------------------------------------------------------------------------
__global__ void __launch_bounds__(THREADS)
iou_partial_kernel(const float* __restrict__ pred,
                   const float* __restrict__ target,
                   const float* __restrict__ weight,
                   float* __restrict__ partials, int n) {
  float acc = 0.0f;
  int stride = gridDim.x * blockDim.x;
  for (int idx = blockIdx.x * blockDim.x + threadIdx.x; idx < n; idx += stride) {
    // hide HBM latency behind the long VALU chain of this iteration
    __builtin_prefetch(pred   + 7 * (idx + stride), 0, 1);
    __builtin_prefetch(target + 7 * (idx + stride), 0, 1);
    acc += pair_loss(pred + 7 * idx, target + 7 * idx, weight[idx]);
  }
  // all lanes reconverged here -> EXEC all ones, WMMA legal
  float wsum = wave_sum32(acc);
  __shared__ float smem[THREADS / 32];
  int lane = threadIdx.x & 31;
  int wid  = threadIdx.x >> 5;
  if (lane == 0) smem[wid] = wsum;
  __syncthreads();
  if (threadIdx.x == 0) {
    float s = 0.0f;
#pragma unroll
    for (int i = 0; i < THREADS / 32; ++i) s += smem[i];
    partials[blockIdx.x] = s;
  }
}

// ---------------------------------------------------------------------------
// Pass 2: reduce NBLOCKS partials, scale by 1/N, write scalar loss
// ---------------------------------------------------------------------------
__global__ void __launch_bounds__(THREADS)
finalize_kernel(const float* __restrict__ partials,
                float* __restrict__ out, int nparts, float scale) {
  float acc = 0.0f;
  for (int i = threadIdx.x; i < nparts; i += blockDim.x) acc += partials[i];
  float wsum = wave_sum32(acc);
  __shared__ float smem[THREADS / 32];
  int lane = threadIdx.x & 31;
  int wid  = threadIdx.x >> 5;
  if (lane == 0) smem[wid] = wsum;
  __syncthreads();
  if (threadIdx.x == 0) {
    float s = 0.0f;
#pragma unroll
    for (int i = 0; i < THREADS / 32; ++i) s += smem[i];
    out[0] = s * scale;   // LOSS_WEIGHT == 1.0
  }
}

extern "C" void kernel_launch(void* const* d_in, const int* in_sizes, int n_in,
                              void* d_out, int out_size, void* d_ws, size_t ws_size,
                              hipStream_t stream) {
  const float* pred   = (const float*)d_in[0];
  const float* target = (const float*)d_in[1];
  const float* weight = (const float*)d_in[2];
  float* out      = (float*)d_out;
  float* partials = (float*)d_ws;            // NBLOCKS floats (4 KB) of scratch
  int n = in_sizes[2];                       // weight has one entry per box

  iou_partial_kernel<<<NBLOCKS, THREADS, 0, stream>>>(pred, target, weight,
                                                      partials, n);
  finalize_kernel<<<1, THREADS, 0, stream>>>(partials, out, NBLOCKS,
                                             1.0f / (float)n);
}